// FFTAgc_60825326846432
// MI455X (gfx1250) — compile-verified
//
#include <hip/hip_runtime.h>
#include <stdint.h>

typedef float        v2f   __attribute__((ext_vector_type(2)));
typedef float        v8f   __attribute__((ext_vector_type(8)));
typedef unsigned int u32x4 __attribute__((ext_vector_type(4)));
typedef int          i32x4 __attribute__((ext_vector_type(4)));
typedef int          i32x8 __attribute__((ext_vector_type(8)));

#define NB        32
#define NT        4000
#define NFRAMES   (NB * NT)          // 128000 frames
#define FRAME_F   322                // 161 bins * 2 (re,im) floats per frame
#define NSPEC     ((size_t)NFRAMES * FRAME_F)   // 41,216,000 floats
#define N2        (NSPEC / 2)        // float2 count = 20,608,000
#define FPB       32                 // frames per block in kernel 1 (2 waves * 16)

// ---------------------------------------------------------------------------
// Kernel 1: per-frame weighted power + sqrt  ->  g0[b*NT+t]
//   TDM stages 32 frames (41,216 B) into LDS, then each wave reduces 16
//   frames with a chain of 80 V_WMMA_F32_16X16X4_F32 (A = x^2, B = ones).
// ---------------------------------------------------------------------------
__global__ __launch_bounds__(64) void k_frame_gain(const float* __restrict__ spec,
                                                   float* __restrict__ g0out)
{
    __shared__ float lds[FPB * FRAME_F];            // 41,216 bytes
    const unsigned tid  = threadIdx.x;
    const unsigned wave = tid >> 5;
    const unsigned lane = tid & 31;

    // ---- TDM: one 1-D tile of 10304 f32 elements, issued by wave 0 ----
    if (wave == 0) {
        const uint64_t ga  = (uint64_t)(const void*)(spec + (size_t)blockIdx.x * FPB * FRAME_F);
        const unsigned la  = (unsigned)(uintptr_t)(void*)lds;   // low 32 bits = LDS offset
        const unsigned ne  = FPB * FRAME_F;                     // 10304 elements

        u32x4 d0;
        d0[0] = 1u;                                  // count=1, user descriptor
        d0[1] = la;                                  // lds_addr
        d0[2] = (unsigned)(ga & 0xFFFFFFFFu);        // global_addr[31:0]
        d0[3] = (unsigned)((ga >> 32) & 0x01FFFFFFu) // global_addr[56:32]
              | (2u << 30);                          // type = 2 ("image")

        i32x8 d1;
        d1[0] = (int)(2u << 16);                     // data_size = 4 bytes
        d1[1] = (int)((ne & 0xFFFFu) << 16);         // tensor_dim0[15:0]  @ bits[63:48]
        d1[2] = (int)((ne >> 16) | (1u << 16));      // tensor_dim0[31:16], tensor_dim1 = 1
        d1[3] = (int)((ne & 0xFFFFu) << 16);         // tile_dim0 @ bits[127:112]
        d1[4] = 0;                                   // tile_dim1 = 0 (1-D), tile_dim2 = 0
        d1[5] = (int)ne;                             // tensor_dim0_stride[31:0]
        d1[6] = 0;
        d1[7] = 0;

        i32x4 z4 = {0, 0, 0, 0};
#if defined(__clang_major__) && (__clang_major__ >= 23)
        i32x8 z8 = {0, 0, 0, 0, 0, 0, 0, 0};
        __builtin_amdgcn_tensor_load_to_lds(d0, d1, z4, z4, z8, 0);
#else
        __builtin_amdgcn_tensor_load_to_lds(d0, d1, z4, z4, 0);
#endif
        __builtin_amdgcn_s_wait_tensorcnt(0);
    }
    __syncthreads();

    // ---- WMMA reduction: S_m = sum_{i=0..319} x_m[i]^2 for 16 frames/wave ----
    // A-layout (32-bit A 16x4): lane L -> M = L&15, K = (L<16 ? 0 : 2) + vgpr.
    const float* W  = lds + (size_t)wave * 16 * FRAME_F;
    const unsigned m  = lane & 15;
    const unsigned kh = lane >> 4;                   // 0 or 1
    const float* rowp = W + (size_t)m * FRAME_F + 2u * kh;

    v8f acc   = {0.f, 0.f, 0.f, 0.f, 0.f, 0.f, 0.f, 0.f};
    v2f bones = {1.0f, 1.0f};                        // B = all-ones 4x16

    #pragma unroll 4
    for (int j = 0; j < 80; ++j) {                   // 80 * K4 = elements 0..319
        v2f x = *(const v2f*)(rowp + 4 * j);
        v2f a; a.x = x.x * x.x; a.y = x.y * x.y;
        acc = __builtin_amdgcn_wmma_f32_16x16x4_f32(
                  false, a, false, bones, (short)0, acc, false, false);
    }

    // D layout: lane L holds M = r + 8*(L>=16) in vgpr r; use N==0 lanes (0,16).
    if (m == 0) {
        const size_t frame0 = (size_t)blockIdx.x * FPB + wave * 16 + kh * 8;
        #pragma unroll
        for (int r = 0; r < 8; ++r) {
            const float* fr = W + (size_t)(kh * 8 + r) * FRAME_F;
            float s    = acc[r];
            // weighted sum = 2*S - x0^2 - x1^2 + x320^2 + x321^2
            float corr = fr[320] * fr[320] + fr[321] * fr[321]
                       - fr[0]   * fr[0]   - fr[1]   * fr[1];
            float p = fmaf(2.0f, s, corr) * (1.0f / 320.0f);
            g0out[frame0 + r] = sqrtf(p);
        }
    }
}

// ---------------------------------------------------------------------------
// Kernel 2: EMA over time, one wave per batch, in place on the gain buffer.
//   c_0 = g0_0 ; c_t = 0.9 c_{t-1} + 0.1 g0_t.  Affine-composition scan.
// ---------------------------------------------------------------------------
__global__ __launch_bounds__(32) void k_ema(float* __restrict__ g)
{
    __shared__ float lds[NT];
    const int lane = threadIdx.x;
    float* gb = g + (size_t)blockIdx.x * NT;

    for (int i = lane; i < NT; i += 32) lds[i] = gb[i];
    __syncthreads();

    const int base = lane * 125;                     // 32 * 125 = 4000
    float A = 1.0f, B = 0.0f;                        // c_out = A*c_in + B
    for (int i = 0; i < 125; ++i) {
        float x = lds[base + i];
        if (base + i == 0) { A = 0.0f; B = x; }      // anchor: c_0 = x_0
        else               { B = fmaf(0.9f, B, 0.1f * x); A *= 0.9f; }
    }
    // Hillis-Steele inclusive scan of compositions across the wave
    for (int d = 1; d < 32; d <<= 1) {
        float pA = __shfl_up(A, d, 32);
        float pB = __shfl_up(B, d, 32);
        if (lane >= d) { B = fmaf(A, pB, B); A = A * pA; }
    }
    float carry = __shfl_up(B, 1, 32);               // exclusive prefix value
    if (lane == 0) carry = 0.0f;

    float c = carry;
    for (int i = 0; i < 125; ++i) {
        float x = lds[base + i];
        c = (base + i == 0) ? x : fmaf(0.9f, c, 0.1f * x);
        gb[base + i] = c;
    }
}

// ---------------------------------------------------------------------------
// Kernel 3: new_spec = spec / (gain + 1e-3), one float2 per thread.
// ---------------------------------------------------------------------------
__global__ __launch_bounds__(256) void k_norm(const float* __restrict__ spec,
                                              const float* __restrict__ gain,
                                              float* __restrict__ out)
{
    size_t i = (size_t)blockIdx.x * 256 + threadIdx.x;
    if (i >= N2) return;
    v2f x = *(const v2f*)(spec + 2 * i);
    unsigned f = (unsigned)i / 161u;                 // 161 float2 per frame
    float inv = 1.0f / (gain[f] + 0.001f);
    v2f y; y.x = x.x * inv; y.y = x.y * inv;
    *(v2f*)(out + 2 * i) = y;
}

// ---------------------------------------------------------------------------
extern "C" void kernel_launch(void* const* d_in, const int* in_sizes, int n_in,
                              void* d_out, int out_size, void* d_ws, size_t ws_size,
                              hipStream_t stream)
{
    (void)in_sizes; (void)n_in; (void)out_size; (void)d_ws; (void)ws_size;
    const float* spec = (const float*)d_in[0];
    float* out  = (float*)d_out;
    float* gbuf = out + NSPEC;                       // gain region of d_out

    k_frame_gain<<<NFRAMES / FPB, 64, 0, stream>>>(spec, gbuf);   // 4000 blocks
    k_ema       <<<NB, 32, 0, stream>>>(gbuf);                    // 32 waves
    k_norm      <<<(unsigned)((N2 + 255) / 256), 256, 0, stream>>>(spec, gbuf, out);
}